// SVDAlign_18657337933934
// MI455X (gfx1250) — compile-verified
//
#include <hip/hip_runtime.h>

typedef __attribute__((ext_vector_type(2))) float v2f;
typedef __attribute__((ext_vector_type(8))) float v8f;
typedef __attribute__((ext_vector_type(4))) int   v4i;
typedef __attribute__((address_space(3)))  v4i    lds_v4i;

#define NEIGH      100
#define ROWF       13
#define BATCH_F    (NEIGH * ROWF)    // 1300 floats per batch
#define OUT_F      (NEIGH * 6)       // 600 floats per batch
#define PRE_STRIDE 7                 // [w*v2x, w*v2y, w*v2z, w, v1x, v1y, v1z]
#define UNI_F      (NEIGH * PRE_STRIDE) // 700 floats: union(pre tile, out tile)
#define WPB        8                 // waves (=batches) per block
#define TPB        (WPB * 32)

// gfx1250 async LDS<->global DMA path, guarded so absence degrades gracefully.
#if defined(__HIP_DEVICE_COMPILE__) && \
    __has_builtin(__builtin_amdgcn_global_load_async_to_lds_b128) && \
    __has_builtin(__builtin_amdgcn_s_wait_asynccnt)
#define ASYNC_LD 1
#else
#define ASYNC_LD 0
#endif
#if defined(__HIP_DEVICE_COMPILE__) && \
    __has_builtin(__builtin_amdgcn_global_store_async_from_lds_b128) && \
    __has_builtin(__builtin_amdgcn_s_wait_asynccnt)
#define ASYNC_ST 1
#else
#define ASYNC_ST 0
#endif

// Global-address arg: generic v4i* (exact type from the compiler diagnostic).
// LDS arg: AS(3) v4i* — matches an AS(3) param exactly, or implicitly widens
// to a generic param.
#define GPTR(p) ((v4i*)(p))
#define LPTR(p) ((lds_v4i*)(p))

__global__ __launch_bounds__(TPB)
void svd_align_kernel(const float* __restrict__ gin,
                      const float* __restrict__ pa,
                      const float* __restrict__ pb,
                      const float* __restrict__ pos,
                      float* __restrict__ gout,
                      int Btot)
{
    __shared__ __align__(16) float sin_[WPB][BATCH_F];
    __shared__ __align__(16) float uni[WPB][UNI_F];

    const int wave = threadIdx.x >> 5;
    const int lane = threadIdx.x & 31;
    const int b    = blockIdx.x * WPB + wave;
    const bool valid = (b < Btot);

    const float A  = pa[0];
    const float Bc = pb[0];
    const float OS = pos[0];

    // ---- stage input tile to LDS ----
    if (valid) {
#if ASYNC_LD
        const char* gsrc = (const char*)(gin + (size_t)b * BATCH_F);
        for (int i = lane; i < BATCH_F / 4; i += 32)
            __builtin_amdgcn_global_load_async_to_lds_b128(
                GPTR(gsrc + (size_t)i * 16), LPTR(&sin_[wave][i * 4]), 0, 0);
        __builtin_amdgcn_s_wait_asynccnt(0);
#else
        const float4* src = (const float4*)(gin + (size_t)b * BATCH_F);
        float4* dst = (float4*)sin_[wave];
        for (int i = lane; i < BATCH_F / 4; i += 32) dst[i] = src[i];
#endif
    }
    __syncthreads();

    if (valid) {
        const float* base = sin_[wave];
        float* preW = uni[wave];

        // ---- per-neighbor precompute tile (branch-free consumers) ----
        for (int n = lane; n < NEIGH; n += 32) {
            const float* r = base + n * ROWF;
            float w = fmaxf(fmaf(A, r[0], Bc), 0.0f) + 1e-8f;
            float* p = preW + n * PRE_STRIDE;
            p[3] = w;
            #pragma unroll
            for (int c = 0; c < 3; ++c) {
                p[c]     = w * fmaf(r[10 + c], OS, r[4 + c]);  // w * v2[c]
                p[4 + c] =     fmaf(r[7 + c],  OS, r[1 + c]);  // v1[c]
            }
        }
        __syncthreads();

        // ---- 16x16x4 FP32 WMMA chain: one instruction stream yields all 16 stats ----
        // A rows: [w*v2x, w*v2y, w*v2z, w, 0...] ; B cols: [v1x, v1y, v1z, 1, 0...]
        const int half = lane >> 4;    // K pair select
        const int l16  = lane & 15;    // A row / B col
        const int ia   = l16 & 3;
        const int ib   = 4 + ((l16 < 3) ? l16 : 0);
        const bool sA  = (l16 < 4);
        const bool sB  = (l16 < 3);
        const float cB = (l16 == 3) ? 1.0f : 0.0f;

        v8f acc = {};
        for (int g = 0; g < NEIGH / 4; ++g) {
            const float* p0 = preW + (4 * g + 2 * half) * PRE_STRIDE;
            const float* p1 = p0 + PRE_STRIDE;
            float a0 = p0[ia], a1 = p1[ia];
            float b0 = p0[ib], b1 = p1[ib];
            v2f af, bf;
            af.x = sA ? a0 : 0.0f;
            af.y = sA ? a1 : 0.0f;
            bf.x = sB ? b0 : cB;
            bf.y = sB ? b1 : cB;
            acc = __builtin_amdgcn_wmma_f32_16x16x4_f32(
                false, af, false, bf, (short)0, acc, false, false);
        }

        // ---- extract stats: D(i,j) = VGPR i, lane j (i,j in 0..3) ----
        float Wsum = __shfl(acc[3], 3);
        float S1[3], S2[3], M[3][3];
        #pragma unroll
        for (int j = 0; j < 3; ++j) S1[j] = __shfl(acc[3], j);
        #pragma unroll
        for (int i = 0; i < 3; ++i) {
            S2[i] = __shfl(acc[i], 3);
            #pragma unroll
            for (int j = 0; j < 3; ++j) M[i][j] = __shfl(acc[i], j);
        }

        const float invW = 1.0f / Wsum;
        float c1[3], c2[3], C[3][3];
        #pragma unroll
        for (int i = 0; i < 3; ++i) { c1[i] = S1[i] * invW; c2[i] = S2[i] * invW; }
        #pragma unroll
        for (int i = 0; i < 3; ++i)
            #pragma unroll
            for (int j = 0; j < 3; ++j)
                C[i][j] = M[i][j] - S2[i] * S1[j] * invW;

        // ---- Jacobi eigendecomposition of C^T C (V = right singular vectors) ----
        float S[3][3], V[3][3], e[3];
        #pragma unroll
        for (int i = 0; i < 3; ++i)
            #pragma unroll
            for (int j = 0; j < 3; ++j) {
                float s = 0.f;
                #pragma unroll
                for (int k = 0; k < 3; ++k) s += C[k][i] * C[k][j];
                S[i][j] = s;
                V[i][j] = (i == j) ? 1.f : 0.f;
            }
        for (int sweep = 0; sweep < 4; ++sweep) {
            const int ps[3] = {0, 0, 1}, qs[3] = {1, 2, 2};
            #pragma unroll
            for (int k = 0; k < 3; ++k) {
                const int p = ps[k], q = qs[k];
                float apq = S[p][q];
                float cth = 1.f, sth = 0.f;
                if (fabsf(apq) > 1e-30f) {
                    float tau = (S[q][q] - S[p][p]) / (2.0f * apq);
                    float t = copysignf(1.0f, tau) /
                              (fabsf(tau) + sqrtf(fmaf(tau, tau, 1.0f)));
                    cth = rsqrtf(fmaf(t, t, 1.0f));
                    sth = t * cth;
                }
                #pragma unroll
                for (int r = 0; r < 3; ++r) {
                    float srp = S[r][p], srq = S[r][q];
                    S[r][p] = cth * srp - sth * srq;
                    S[r][q] = sth * srp + cth * srq;
                }
                #pragma unroll
                for (int r = 0; r < 3; ++r) {
                    float spr = S[p][r], sqr = S[q][r];
                    S[p][r] = cth * spr - sth * sqr;
                    S[q][r] = sth * spr + cth * sqr;
                }
                #pragma unroll
                for (int r = 0; r < 3; ++r) {
                    float vrp = V[r][p], vrq = V[r][q];
                    V[r][p] = cth * vrp - sth * vrq;
                    V[r][q] = sth * vrp + cth * vrq;
                }
            }
        }
        e[0] = S[0][0]; e[1] = S[1][1]; e[2] = S[2][2];

        // sort eigenpairs descending (column 2 = smallest singular value)
        #pragma unroll
        for (int a0_ = 0; a0_ < 2; ++a0_)
            #pragma unroll
            for (int b0_ = 0; b0_ < 2 - a0_; ++b0_) {
                if (e[b0_] < e[b0_ + 1]) {
                    float te = e[b0_]; e[b0_] = e[b0_ + 1]; e[b0_ + 1] = te;
                    #pragma unroll
                    for (int r = 0; r < 3; ++r) {
                        float tv = V[r][b0_]; V[r][b0_] = V[r][b0_ + 1]; V[r][b0_ + 1] = tv;
                    }
                }
            }

        // ---- left singular vectors: u_k = normalize(C*v_k), u2 = u0 x u1 ----
        float u0[3], u1[3], u2[3];
        #pragma unroll
        for (int i = 0; i < 3; ++i)
            u0[i] = C[i][0] * V[0][0] + C[i][1] * V[1][0] + C[i][2] * V[2][0];
        float n0s = u0[0]*u0[0] + u0[1]*u0[1] + u0[2]*u0[2];
        float in0 = rsqrtf(fmaxf(n0s, 1e-30f));
        #pragma unroll
        for (int i = 0; i < 3; ++i) u0[i] *= in0;

        #pragma unroll
        for (int i = 0; i < 3; ++i)
            u1[i] = C[i][0] * V[0][1] + C[i][1] * V[1][1] + C[i][2] * V[2][1];
        float d01 = u1[0]*u0[0] + u1[1]*u0[1] + u1[2]*u0[2];
        #pragma unroll
        for (int i = 0; i < 3; ++i) u1[i] -= d01 * u0[i];
        float n1s = u1[0]*u1[0] + u1[1]*u1[1] + u1[2]*u1[2];
        float in1 = rsqrtf(fmaxf(n1s, 1e-30f));
        #pragma unroll
        for (int i = 0; i < 3; ++i) u1[i] *= in1;

        u2[0] = u0[1] * u1[2] - u0[2] * u1[1];
        u2[1] = u0[2] * u1[0] - u0[0] * u1[2];
        u2[2] = u0[0] * u1[1] - u0[1] * u1[0];

        float detV = V[0][0] * (V[1][1] * V[2][2] - V[1][2] * V[2][1])
                   - V[0][1] * (V[1][0] * V[2][2] - V[1][2] * V[2][0])
                   + V[0][2] * (V[1][0] * V[2][1] - V[1][1] * V[2][0]);

        // R = V * diag(1,1,detV) * U^T ; t = c1 - R*c2
        float R[3][3], T[3];
        #pragma unroll
        for (int i = 0; i < 3; ++i)
            #pragma unroll
            for (int j = 0; j < 3; ++j)
                R[i][j] = V[i][0] * u0[j] + V[i][1] * u1[j] + detV * V[i][2] * u2[j];
        #pragma unroll
        for (int i = 0; i < 3; ++i)
            T[i] = c1[i] - (R[i][0] * c2[0] + R[i][1] * c2[1] + R[i][2] * c2[2]);

        // ---- transform all neighbors into LDS out tile (overwrites pre tile) ----
        float* soutW = uni[wave];
        for (int n = lane; n < NEIGH; n += 32) {
            const float* r = base + n * ROWF;
            float x  = r[4],  y  = r[5],  z  = r[6];
            float nx = r[10], ny = r[11], nz = r[12];
            float* o = soutW + n * 6;
            o[0] = R[0][0]*x + R[0][1]*y + R[0][2]*z + T[0];
            o[1] = R[1][0]*x + R[1][1]*y + R[1][2]*z + T[1];
            o[2] = R[2][0]*x + R[2][1]*y + R[2][2]*z + T[2];
            o[3] = R[0][0]*nx + R[0][1]*ny + R[0][2]*nz;
            o[4] = R[1][0]*nx + R[1][1]*ny + R[1][2]*nz;
            o[5] = R[2][0]*nx + R[2][1]*ny + R[2][2]*nz;
        }
    }
    __syncthreads();   // s_wait_dscnt 0 + barrier: out tile complete in LDS

    // ---- coalesced 128b drain of out tile ----
    if (valid) {
#if ASYNC_ST
        char* gdst = (char*)(gout + (size_t)b * OUT_F);
        for (int i = lane; i < OUT_F / 4; i += 32)
            __builtin_amdgcn_global_store_async_from_lds_b128(
                GPTR(gdst + (size_t)i * 16), LPTR(&uni[wave][i * 4]), 0, 0);
#else
        float4* dst = (float4*)(gout + (size_t)b * OUT_F);
        const float4* src = (const float4*)uni[wave];
        for (int i = lane; i < OUT_F / 4; i += 32) dst[i] = src[i];
#endif
    }
}

extern "C" void kernel_launch(void* const* d_in, const int* in_sizes, int n_in,
                              void* d_out, int out_size, void* d_ws, size_t ws_size,
                              hipStream_t stream) {
    const float* gin = (const float*)d_in[0];
    const float* pa  = (const float*)d_in[1];
    const float* pb  = (const float*)d_in[2];
    const float* pos = (const float*)d_in[3];
    float* gout = (float*)d_out;

    const int Btot = in_sizes[0] / BATCH_F;
    const int blocks = (Btot + WPB - 1) / WPB;
    svd_align_kernel<<<blocks, TPB, 0, stream>>>(gin, pa, pb, pos, gout, Btot);
}